// MultiScaleDCN_31533649887721
// MI455X (gfx1250) — compile-verified
//
#include <hip/hip_runtime.h>

// Problem constants (from reference): B,H,W,G,C,K = 4,128,128,8,32,8
#define Bn 4
#define Hn 128
#define Wn 128
#define Gn 8
#define Cn 32
#define Kn 8
#define PG_PER_BLOCK 8          // 8 waves of 32 lanes = 256 threads; 1 wave per (b,h,w,g)

// LDS: deformables region [0..127] (pg*16+i), weights region [128..191] (128+pg*8+k)
#define DEF_DWORDS (PG_PER_BLOCK * 2 * Kn)   // 128
#define WTS_DWORDS (PG_PER_BLOCK * Kn)       // 64

typedef unsigned int v4u __attribute__((ext_vector_type(4)));
typedef int          v8i __attribute__((ext_vector_type(8)));
typedef int          v4i __attribute__((ext_vector_type(4)));

__global__ __launch_bounds__(256) void MultiScaleDCN_kernel(
    const float* __restrict__ inputs,      // [B,H,W,G,C]
    const float* __restrict__ deformables, // [B,H,W,G,K,2]
    const float* __restrict__ weights,     // [B,H,W,G,K]
    float* __restrict__ out)               // [B,H,W,G,C]
{
    __shared__ float s_meta[DEF_DWORDS + WTS_DWORDS];

    const int tid     = threadIdx.x;
    const int pg_base = blockIdx.x * PG_PER_BLOCK;

    // ---- Weights tile (256B) via per-lane gfx1250 async copies (ASYNCcnt path).
    if (tid < WTS_DWORDS) {
        unsigned lds_dst = (unsigned)(size_t)&s_meta[DEF_DWORDS + tid];
        unsigned goff    = (unsigned)((pg_base * Kn + tid) * 4);
        asm volatile("global_load_async_to_lds_b32 %0, %1, %2"
                     :: "v"(lds_dst), "v"(goff), "s"(weights)
                     : "memory");
    }

    // ---- Deformables tile (512B) via the Tensor Data Mover: one descriptor,
    // ---- issued once by wave 0 (TDM ignores EXEC; all operands are scalar),
    // ---- tracked on TENSORcnt independently of the ASYNCcnt weights copy.
    if (tid < 32) {
        const unsigned lds_addr = (unsigned)(size_t)&s_meta[0];     // LDS byte offset
        const unsigned long long ga =
            (unsigned long long)(const void*)(deformables + (size_t)pg_base * 2 * Kn);
        const unsigned N = DEF_DWORDS;                               // 128 dwords

        // D# group 0: count=1 | lds_addr | global_addr[56:0] | type=2 (ISA 8.3)
        v4u g0 = { 1u,
                   lds_addr,
                   (unsigned)ga,
                   (unsigned)((ga >> 32) & 0x01FFFFFFu) | (2u << 30) };
        // D# group 1: data_size=4B; tensor_dim0=N, tensor_dim1=1;
        //             tile_dim0=N, tile_dim1=1; tensor_dim0_stride=N (ISA 8.4)
        v8i g1 = { (int)0x00020000u,                 // workgroup_mask=0, data_size=2(4B)
                   (int)((N & 0xFFFFu) << 16),       // tensor_dim0[15:0]
                   (int)((N >> 16) | (1u << 16)),    // tensor_dim0[31:16] | tensor_dim1[15:0]
                   (int)(N << 16),                   // tensor_dim1[31:16] | tile_dim0
                   (int)1,                           // tile_dim1=1, tile_dim2=0
                   (int)N,                           // tensor_dim0_stride[31:0]
                   0, 0 };
        v4i z4 = { 0, 0, 0, 0 };                     // groups 2/3 unused (<=2D tensor)
#if __clang_major__ >= 23
        v8i z8 = { 0, 0, 0, 0, 0, 0, 0, 0 };
        __builtin_amdgcn_tensor_load_to_lds(g0, g1, z4, z4, z8, 0);
#else
        __builtin_amdgcn_tensor_load_to_lds(g0, g1, z4, z4, 0);
#endif
        __builtin_amdgcn_s_wait_tensorcnt((short)0); // wave 0: TDM complete
    }
    asm volatile("s_wait_asynccnt 0" ::: "memory");  // per-wave async copies complete
    __syncthreads();

    // ---- One wave per pixel-group, lane = channel (wave32 == C lanes exactly).
    // pg_local is wave-uniform: readfirstlane pins it (and everything derived
    // from it) into SGPRs so coordinate math runs on the SALU pipe.
    const int pg_local = __builtin_amdgcn_readfirstlane(tid >> 5);
    const int lane     = tid & 31;
    const int pg       = pg_base + pg_local;                   // scalar
    const float* mpd   = &s_meta[pg_local * 2 * Kn];           // 16 deformable floats
    const float* mpw   = &s_meta[DEF_DWORDS + pg_local * Kn];  // 8 weight floats

    // pg = ((b*H + h)*W + w)*G + g  with G=8, W=128, H=128
    const int g  = pg & (Gn - 1);
    const int t1 = pg >> 3;           // (b*H + h)*W + w
    const int wv = t1 & (Wn - 1);
    const int t2 = t1 >> 7;           // b*H + h
    const int hv = t2 & (Hn - 1);
    const int bv = t2 >> 7;

    // Scalar base for (b, *, *, g); lane adds the channel. Each gather below is
    // saddr-form: scalar (base + corner offset) + per-lane channel offset.
    const float* inb = inputs + (size_t)bv * (Hn * (size_t)Wn * Gn * Cn)
                              + (size_t)g * Cn + lane;
    const int xystride = Gn * Cn;     // 256 floats per x step; row step = Wn*xystride

    float acc = 0.0f;
    const float wf = (float)wv;
    const float hf = (float)hv;

#pragma unroll
    for (int k = 0; k < Kn; ++k) {
        const float xf = mpd[2 * k + 0] + wf;   // LDS broadcast reads (uniform)
        const float yf = mpd[2 * k + 1] + hf;
        const float wk = mpw[k];

        // Truncation toward zero (matches astype(int32) in reference)
        const int fx = (int)xf;
        const int fy = (int)yf;
        const int cx = fx + 1;
        const int cy = fy + 1;
        const float fxf = (float)fx, fyf = (float)fy;
        const float cxf = (float)cx, cyf = (float)cy;

        const float wx0 = cxf - xf, wx1 = xf - fxf;
        const float wy0 = cyf - yf, wy1 = yf - fyf;

        // Masks on UNCLAMPED indices, gathers on CLAMPED indices (matches reference)
        const bool mx0 = (fx >= 0) && (fx < Wn);
        const bool my0 = (fy >= 0) && (fy < Hn);
        const bool mx1 = (cx >= 0) && (cx < Wn);
        const bool my1 = (cy >= 0) && (cy < Hn);

        const int fxc = min(max(fx, 0), Wn - 1);
        const int fyc = min(max(fy, 0), Hn - 1);
        const int cxc = min(max(cx, 0), Wn - 1);
        const int cyc = min(max(cy, 0), Hn - 1);

        // Wave-uniform flat element offsets -> force into SGPRs so the four
        // gathers become saddr-form global_load_b32 (one shared lane VGPR).
        const int off00 = __builtin_amdgcn_readfirstlane((fyc * Wn + fxc) * xystride);
        const int off01 = __builtin_amdgcn_readfirstlane((fyc * Wn + cxc) * xystride);
        const int off10 = __builtin_amdgcn_readfirstlane((cyc * Wn + fxc) * xystride);
        const int off11 = __builtin_amdgcn_readfirstlane((cyc * Wn + cxc) * xystride);

        // Four coalesced 128B wave gathers (L2-resident: input = 64MiB < 192MB L2)
        const float v00 = inb[off00];
        const float v01 = inb[off01];
        const float v10 = inb[off10];
        const float v11 = inb[off11];

        const float b00 = (mx0 && my0) ? wx0 * wy0 : 0.0f;
        const float b01 = (mx1 && my0) ? wx1 * wy0 : 0.0f;
        const float b10 = (mx0 && my1) ? wx0 * wy1 : 0.0f;
        const float b11 = (mx1 && my1) ? wx1 * wy1 : 0.0f;

        float s = v00 * b00;
        s = fmaf(v01, b01, s);
        s = fmaf(v10, b10, s);
        s = fmaf(v11, b11, s);
        acc = fmaf(wk, s, acc);
    }

    // Output written once, never re-read: non-temporal so it doesn't evict the
    // L2-resident input tensor.
    __builtin_nontemporal_store(acc, out + (size_t)pg * Cn + lane);
}

extern "C" void kernel_launch(void* const* d_in, const int* in_sizes, int n_in,
                              void* d_out, int out_size, void* d_ws, size_t ws_size,
                              hipStream_t stream) {
    const float* inputs      = (const float*)d_in[0];
    const float* deformables = (const float*)d_in[1];
    const float* weights     = (const float*)d_in[2];
    float* out               = (float*)d_out;

    const int total_pg = Bn * Hn * Wn * Gn;            // 524288
    dim3 grid(total_pg / PG_PER_BLOCK);                // 65536 blocks
    dim3 block(256);                                   // 8 wave32 waves
    MultiScaleDCN_kernel<<<grid, block, 0, stream>>>(inputs, deformables, weights, out);
}